// Net_25675314495976
// MI455X (gfx1250) — compile-verified
//
#include <hip/hip_runtime.h>
#include <hip/hip_bf16.h>
#include <stdint.h>

typedef __attribute__((ext_vector_type(2))) float v2f;
typedef __attribute__((ext_vector_type(8))) float v8f;

// ---------------------------------------------------------------------------
// degree / norm kernels
// ---------------------------------------------------------------------------
__global__ void k_deg_init(float* __restrict__ deg, int n) {
    int i = blockIdx.x * blockDim.x + threadIdx.x;
    if (i < n) deg[i] = 1.0f;  // self-loop contributes 1 to degree
}

__global__ void k_deg_acc(const long long* __restrict__ dst,
                          float* __restrict__ deg, unsigned int E) {
    unsigned int e = blockIdx.x * blockDim.x + threadIdx.x;
    if (e < E) atomicAdd(&deg[(int)dst[e]], 1.0f);
}

__global__ void k_rsqrt_inplace(float* __restrict__ deg, int n) {
    int i = blockIdx.x * blockDim.x + threadIdx.x;
    if (i < n) deg[i] = rsqrtf(deg[i]);
}

__global__ void k_zero(float* __restrict__ p, unsigned int n) {
    unsigned int i = blockIdx.x * blockDim.x + threadIdx.x;
    if (i < n) p[i] = 0.0f;
}

// ---------------------------------------------------------------------------
// WMMA f32 GEMM: H[nrows x NDIM] = X[nrows x KDIM] @ W[KDIM x NDIM]
// One wave (32 lanes) computes a 16-row tile via V_WMMA_F32_16X16X4_F32.
// A 16x4 layout: lanes 0-15 row M=lane hold K=0(v0),1(v1); lanes 16-31 same
// rows hold K=2(v0),3(v1).  B 4x16: v0 = rows K=0 (lanes 0-15) / K=2
// (lanes 16-31), v1 = rows K=1 / K=3.  C/D: VGPR g, lanes 0-15 -> M=g,
// lanes 16-31 -> M=8+g, N = lane % 16.
// Store path is tile-uniform: full tiles take an unguarded 8-store clause
// (EXEC stays all-ones, no per-row branches); ragged tail is the slow path.
// ---------------------------------------------------------------------------
template <int KDIM, int NDIM>
__global__ void k_gemm_wmma(const float* __restrict__ X,
                            const float* __restrict__ W,
                            float* __restrict__ H, int nrows) {
    constexpr int NCHUNK = KDIM / 4;            // K in steps of 4
    constexpr int NTILE  = (NDIM + 15) / 16;    // N in tiles of 16
    const int wave = (int)((blockIdx.x * blockDim.x + threadIdx.x) >> 5);
    const int lane = threadIdx.x & 31;
    const int row0 = wave * 16;
    if (row0 >= nrows) return;                  // uniform per wave
    const bool full_tile = (row0 + 16 <= nrows); // uniform per wave

    const int mrow = full_tile ? (row0 + (lane & 15))
                               : min(row0 + (lane & 15), nrows - 1);
    const int koff = (lane >> 4) * 2;           // 0 for lanes 0-15, 2 for 16-31

    v2f afrag[NCHUNK];
#pragma unroll
    for (int c = 0; c < NCHUNK; ++c) {
        const float* p = X + (size_t)mrow * KDIM + c * 4 + koff;
        afrag[c].x = p[0];
        afrag[c].y = p[1];
    }

#pragma unroll
    for (int t = 0; t < NTILE; ++t) {
        const int ncol = t * 16 + (lane & 15);
        v8f acc = {};
#pragma unroll
        for (int c = 0; c < NCHUNK; ++c) {
            v2f b;
            if (ncol < NDIM) {
                b.x = W[(c * 4 + koff) * NDIM + ncol];
                b.y = W[(c * 4 + koff + 1) * NDIM + ncol];
            } else {
                b.x = 0.0f; b.y = 0.0f;
            }
            acc = __builtin_amdgcn_wmma_f32_16x16x4_f32(
                /*neg_a=*/false, afrag[c], /*neg_b=*/false, b,
                /*c_mod=*/(short)0, acc, /*reuse_a=*/false, /*reuse_b=*/false);
        }
        const int rbase = row0 + 8 * (lane >> 4);
        float* hp = H + (size_t)rbase * NDIM + ncol;
        if (full_tile) {
            if (ncol < NDIM) {
#pragma unroll
                for (int g = 0; g < 8; ++g) hp[(size_t)g * NDIM] = acc[g];
            }
        } else {
            if (ncol < NDIM) {
#pragma unroll
                for (int g = 0; g < 8; ++g) {
                    if (rbase + g < nrows) hp[(size_t)g * NDIM] = acc[g];
                }
            }
        }
    }
}

// ---------------------------------------------------------------------------
// edge scatter: agg[dst, f] += dinv[src]*dinv[dst] * h[src, f]
// one thread per (edge, feature); E*F < 2^31 so 32-bit index math.
// ---------------------------------------------------------------------------
template <int F>
__global__ void k_scatter(const float* __restrict__ h,
                          const long long* __restrict__ src,
                          const long long* __restrict__ dst,
                          const float* __restrict__ dinv,
                          float* __restrict__ agg, unsigned int E) {
    unsigned int idx = blockIdx.x * blockDim.x + threadIdx.x;
    unsigned int total = E * (unsigned int)F;
    if (idx >= total) return;
    unsigned int e = idx / (unsigned int)F;
    unsigned int f = idx - e * (unsigned int)F;
    int s = (int)src[e];
    int d = (int)dst[e];
    float nrm = dinv[s] * dinv[d];
    atomicAdd(&agg[(size_t)d * F + f], nrm * h[(size_t)s * F + f]);
}

// ---------------------------------------------------------------------------
// fused self-loop + bias + optional ReLU: out = act(agg + dinv^2*h + b)
// (self-loop norm is dinv[i]*dinv[i]); safe in place (out == agg).
// ---------------------------------------------------------------------------
template <int F, bool RELU>
__global__ void k_selfloop_bias_act(float* __restrict__ agg,
                                    const float* __restrict__ h,
                                    const float* __restrict__ dinv,
                                    const float* __restrict__ bias, int n) {
    unsigned int idx = blockIdx.x * blockDim.x + threadIdx.x;
    unsigned int total = (unsigned int)n * F;
    if (idx >= total) return;
    unsigned int i = idx / (unsigned int)F;
    unsigned int f = idx - i * (unsigned int)F;
    float di = dinv[i];
    float v = agg[idx] + di * di * h[idx] + bias[f];
    agg[idx] = RELU ? fmaxf(v, 0.0f) : v;
}

// ---------------------------------------------------------------------------
// h3[i] = dot(x2[i, 0:16], W3[0:16])
// ---------------------------------------------------------------------------
__global__ void k_h3(const float* __restrict__ x2,
                     const float* __restrict__ W3,
                     float* __restrict__ h3, int n) {
    int i = blockIdx.x * blockDim.x + threadIdx.x;
    if (i >= n) return;
    const float* row = x2 + (size_t)i * 16;
    float acc = 0.0f;
#pragma unroll
    for (int f = 0; f < 16; ++f) acc = fmaf(row[f], W3[f], acc);
    h3[i] = acc;
}

__global__ void k_out_init(float* __restrict__ out,
                           const float* __restrict__ b3, int n) {
    if (blockIdx.x == 0 && threadIdx.x == 0) out[0] = (float)n * b3[0];
}

// ---------------------------------------------------------------------------
// final reduction: sum_e dinv[s]*dinv[d]*h3[s]  +  sum_i dinv[i]^2*h3[i]
// (last GCN layer followed by global sum collapses to an edge-weighted sum)
// ---------------------------------------------------------------------------
__global__ void k_final_reduce(const long long* __restrict__ src,
                               const long long* __restrict__ dst,
                               const float* __restrict__ dinv,
                               const float* __restrict__ h3,
                               float* __restrict__ out,
                               unsigned int E, unsigned int n) {
    unsigned int idx = blockIdx.x * blockDim.x + threadIdx.x;
    unsigned int total = E + n;
    float v = 0.0f;
    if (idx < E) {
        int s = (int)src[idx];
        int d = (int)dst[idx];
        v = dinv[s] * dinv[d] * h3[s];
    } else if (idx < total) {
        int i = (int)(idx - E);
        v = dinv[i] * dinv[i] * h3[i];
    }
    // wave32 reduction
#pragma unroll
    for (int off = 16; off > 0; off >>= 1) v += __shfl_down(v, off, 32);
    __shared__ float red[8];
    int lane = threadIdx.x & 31;
    int w    = threadIdx.x >> 5;
    if (lane == 0) red[w] = v;
    __syncthreads();
    if (w == 0) {
        v = (lane < (int)(blockDim.x >> 5)) ? red[lane] : 0.0f;
#pragma unroll
        for (int off = 4; off > 0; off >>= 1) v += __shfl_down(v, off, 32);
        if (lane == 0) atomicAdd(out, v);
    }
}

// ---------------------------------------------------------------------------
// launch
// ---------------------------------------------------------------------------
extern "C" void kernel_launch(void* const* d_in, const int* in_sizes, int n_in,
                              void* d_out, int out_size, void* d_ws, size_t ws_size,
                              hipStream_t stream) {
    (void)n_in; (void)out_size; (void)ws_size;

    const float*     x    = (const float*)d_in[0];          // [N,16]
    const long long* ei   = (const long long*)d_in[1];      // [2,E] int64
    const float*     W1   = (const float*)d_in[2];          // [16,20]
    const float*     b1   = (const float*)d_in[3];          // [20]
    const float*     W2   = (const float*)d_in[4];          // [20,16]
    const float*     b2   = (const float*)d_in[5];          // [16]
    const float*     W3   = (const float*)d_in[6];          // [16,1]
    const float*     b3   = (const float*)d_in[7];          // [1]
    float*           out  = (float*)d_out;

    const int          N = in_sizes[0] / 16;                // 100000
    const unsigned int E = (unsigned int)(in_sizes[1] / 2); // 3200000
    const long long* src = ei;
    const long long* dst = ei + E;

    // workspace layout (floats), ~29.6 MB total
    float* ws   = (float*)d_ws;
    float* dinv = ws;                      size_t o = (size_t)N;          // deg -> dinv in place
    float* h1   = ws + o;                  o += (size_t)N * 20;
    float* agg1 = ws + o;                  o += (size_t)N * 20;          // becomes x1 in place
    float* h2   = ws + o;                  o += (size_t)N * 16;
    float* agg2 = ws + o;                  o += (size_t)N * 16;          // becomes x2 in place
    float* h3   = ws + o;                  o += (size_t)N;

    const int B = 256;
    auto cdiv = [](unsigned int a, unsigned int b) { return (a + b - 1) / b; };

    // 1) degrees -> dinv
    k_deg_init<<<cdiv(N, B), B, 0, stream>>>(dinv, N);
    k_deg_acc<<<cdiv(E, B), B, 0, stream>>>(dst, dinv, E);
    k_rsqrt_inplace<<<cdiv(N, B), B, 0, stream>>>(dinv, N);

    // 2) zero accumulators (must happen every call: atomics + poisoned ws)
    k_zero<<<cdiv((unsigned int)N * 20, B), B, 0, stream>>>(agg1, (unsigned int)N * 20);
    k_zero<<<cdiv((unsigned int)N * 16, B), B, 0, stream>>>(agg2, (unsigned int)N * 16);

    // 3) layer 1: h1 = x @ W1 ; scatter ; self-loop + bias + relu
    {
        unsigned int tiles  = cdiv((unsigned int)N, 16);         // one wave each
        unsigned int blocks = cdiv(tiles, (unsigned int)(B / 32));
        k_gemm_wmma<16, 20><<<blocks, B, 0, stream>>>(x, W1, h1, N);
    }
    k_scatter<20><<<cdiv(E * 20u, B), B, 0, stream>>>(h1, src, dst, dinv, agg1, E);
    k_selfloop_bias_act<20, true><<<cdiv((unsigned int)N * 20, B), B, 0, stream>>>(agg1, h1, dinv, b1, N);

    // 4) layer 2: h2 = x1 @ W2 ; scatter ; self-loop + bias + relu
    {
        unsigned int tiles  = cdiv((unsigned int)N, 16);
        unsigned int blocks = cdiv(tiles, (unsigned int)(B / 32));
        k_gemm_wmma<20, 16><<<blocks, B, 0, stream>>>(agg1, W2, h2, N);
    }
    k_scatter<16><<<cdiv(E * 16u, B), B, 0, stream>>>(h2, src, dst, dinv, agg2, E);
    k_selfloop_bias_act<16, true><<<cdiv((unsigned int)N * 16, B), B, 0, stream>>>(agg2, h2, dinv, b2, N);

    // 5) layer 3 collapsed into a weighted global reduction
    k_h3<<<cdiv(N, B), B, 0, stream>>>(agg2, W3, h3, N);
    k_out_init<<<1, 32, 0, stream>>>(out, b3, N);
    k_final_reduce<<<cdiv(E + (unsigned int)N, B), B, 0, stream>>>(src, dst, dinv, h3, out, E, (unsigned int)N);
}